// VmapSphConvAttention_14336600834788
// MI455X (gfx1250) — compile-verified
//
#include <hip/hip_runtime.h>
#include <hip/hip_bf16.h>

#define NHEADS 3
#define FH     64
#define FTOT   192
#define MTOT   15

typedef __attribute__((ext_vector_type(2))) float v2f;
typedef __attribute__((ext_vector_type(8))) float v8f;

// ---------------------------------------------------------------------------
// Kernel 0: zero the output (harness poisons d_out with 0xAA).
// ---------------------------------------------------------------------------
__global__ void zero_out_kernel(float* __restrict__ out, int n) {
    int i = blockIdx.x * blockDim.x + threadIdx.x;
    if (i < n) out[i] = 0.0f;
}

// ---------------------------------------------------------------------------
// Kernel A: per-head Linear projections q = xh @ Wq[h]^T, k = xh @ Wk[h]^T
// using V_WMMA_F32_16X16X4_F32 (fp32 matrix path, matches reference dtype).
//
// One wave32 computes one 16(M nodes) x 16(N out-features) tile, looping
// K=64 in steps of 4 -> 16 v_wmma ops per tile.
//
// A (16x4 f32) VGPR layout: lane L holds row M = L%16; VGPR0/1 hold
// K = (L/16)*2 + {0,1}.   B (4x16) symmetric: lane L holds col N = L%16,
// VGPR0/1 hold K = (L/16)*2 + {0,1}.   C/D: VGPR r holds row r + (L/16)*8,
// col = L%16.
// ---------------------------------------------------------------------------
__global__ __launch_bounds__(256) void qk_project_kernel(
        const float* __restrict__ x,    // (n, 192)
        const float* __restrict__ Wq,   // (3, 64, 64)  row-major (g, f)
        const float* __restrict__ Wk,   // (3, 64, 64)
        float* __restrict__ qout,       // (n, 192)
        float* __restrict__ kout,       // (n, 192)
        int n) {
    const int lane = threadIdx.x;                       // 0..31
    const int ty   = blockIdx.y * blockDim.y + threadIdx.y;  // 0..23
    const int z    = (ty >= 12) ? 1 : 0;                // 0 = q, 1 = k
    const int ct   = ty - z * 12;                       // 0..11
    const int head = ct >> 2;                           // 0..2
    const int nt   = ct & 3;                            // 16-col tile in head

    const int rowTile = blockIdx.x;
    const int mrow = rowTile * 16 + (lane & 15);
    const int mc   = (mrow < n) ? mrow : (n - 1);       // clamp tail (no divergence)
    const int kHalf = (lane >> 4) * 2;                  // 0 or 2
    const int ncol = lane & 15;

    const float* __restrict__ W  = (z == 0) ? Wq : Wk;
    float* __restrict__ out      = (z == 0) ? qout : kout;

    const float* __restrict__ xrow = x + (size_t)mc * FTOT + head * FH;
    // B[k][ncol] = W[head][g = nt*16+ncol][f = k]
    const float* __restrict__ Wcol = W + ((size_t)head * FH + (size_t)(nt * 16 + ncol)) * FH;

    v8f acc = {};
#if __has_builtin(__builtin_amdgcn_wmma_f32_16x16x4_f32)
    #pragma unroll
    for (int kk = 0; kk < FH; kk += 4) {
        v2f a, b;
        a[0] = xrow[kk + kHalf + 0];
        a[1] = xrow[kk + kHalf + 1];
        b[0] = Wcol[kk + kHalf + 0];
        b[1] = Wcol[kk + kHalf + 1];
        // 8 args: (neg_a, A, neg_b, B, c_mod, C, reuse_a, reuse_b)
        acc = __builtin_amdgcn_wmma_f32_16x16x4_f32(
                false, a, false, b, (short)0, acc, false, false);
    }
#else
    // Scalar fallback (should not trigger on gfx1250) — keeps kernel compiling.
    #pragma unroll 4
    for (int kk = 0; kk < FH; ++kk) {
        float av = xrow[kk];
        #pragma unroll
        for (int r = 0; r < 8; ++r) acc[r] += av * Wcol[kk];
    }
#endif

    // Store D: VGPR r -> row (rowTile*16 + r + (lane>>4)*8), col = head*64+nt*16+ncol
    const int colOut = head * FH + nt * 16 + ncol;
    const int rbase  = rowTile * 16 + (lane >> 4) * 8;
    #pragma unroll
    for (int r = 0; r < 8; ++r) {
        int mr = rbase + r;
        if (mr < n) out[(size_t)mr * FTOT + colOut] = acc[r];
    }
}

// ---------------------------------------------------------------------------
// Kernel B: one wave32 per pair.
//   alpha_h = (1/8) * (phi_r[p] + phi_chi[p]) * sum_f q_i[h,f]*w[p,h,f]*k_j[h,f]
//   out[i, m] += alpha_{head(m)} * sph[p, m]   (atomic scatter-add)
// Lane l covers features l, l+32, ..., l+160: fully coalesced; chunks t={0,1}
// belong to head0, {2,3} head1, {4,5} head2 (FH=64 = 2*32).
// ---------------------------------------------------------------------------
__global__ __launch_bounds__(256) void edge_attn_kernel(
        const float* __restrict__ q,       // (n, 192)
        const float* __restrict__ k,       // (n, 192)
        const float* __restrict__ w_ij,    // (P, 192)
        const float* __restrict__ sph,     // (P, 15)
        const float* __restrict__ phi_r,   // (P,)
        const float* __restrict__ phi_c,   // (P,)
        const int*   __restrict__ idx_i,   // (P,)
        const int*   __restrict__ idx_j,   // (P,)
        float* __restrict__ out,           // (n, 15)
        int npairs) {
    const int gtid = blockIdx.x * blockDim.x + threadIdx.x;
    const int p    = gtid >> 5;       // wave id == pair id
    const int lane = threadIdx.x & 31;
    if (p >= npairs) return;          // wave-uniform exit

    const int i = idx_i[p];
    const int j = idx_j[p];
    const float* __restrict__ qi = q    + (size_t)i * FTOT;
    const float* __restrict__ kj = k    + (size_t)j * FTOT;
    const float* __restrict__ wp = w_ij + (size_t)p * FTOT;

    float s0 = 0.f, s1 = 0.f, s2 = 0.f;
    #pragma unroll
    for (int t = 0; t < 6; ++t) {
        const int f = lane + (t << 5);
        const float v = qi[f] * wp[f] * kj[f];
        if (t < 2)      s0 += v;
        else if (t < 4) s1 += v;
        else            s2 += v;
    }
    // wave32 xor-shuffle reduction -> all lanes hold the full head sums
    #pragma unroll
    for (int off = 16; off > 0; off >>= 1) {
        s0 += __shfl_xor(s0, off, 32);
        s1 += __shfl_xor(s1, off, 32);
        s2 += __shfl_xor(s2, off, 32);
    }

    const float scale = 0.125f * (phi_r[p] + phi_c[p]);  // 1/sqrt(64) * (r+chi)
    const float a0 = s0 * scale, a1 = s1 * scale, a2 = s2 * scale;

    if (lane < MTOT) {
        const int m = lane;
        const float ah = (m < 3) ? a0 : ((m < 8) ? a1 : a2);  // repeats [3,5,7]
        atomicAdd(out + (size_t)i * MTOT + m, ah * sph[(size_t)p * MTOT + m]);
    }
}

// ---------------------------------------------------------------------------
// Launch
// inputs: 0 chi (unused), 1 sph_ij, 2 x, 3 w_ij, 4 phi_r_cut, 5 phi_chi_cut,
//         6 Wq, 7 Wk, 8 idx_i, 9 idx_j
// ---------------------------------------------------------------------------
extern "C" void kernel_launch(void* const* d_in, const int* in_sizes, int n_in,
                              void* d_out, int out_size, void* d_ws, size_t ws_size,
                              hipStream_t stream) {
    const float* sph   = (const float*)d_in[1];
    const float* x     = (const float*)d_in[2];
    const float* w_ij  = (const float*)d_in[3];
    const float* phi_r = (const float*)d_in[4];
    const float* phi_c = (const float*)d_in[5];
    const float* Wq    = (const float*)d_in[6];
    const float* Wk    = (const float*)d_in[7];
    const int*   idx_i = (const int*)d_in[8];
    const int*   idx_j = (const int*)d_in[9];
    float* out = (float*)d_out;

    const int n      = in_sizes[2] / FTOT;   // 25000
    const int npairs = in_sizes[8];          // 400000

    float* qbuf = (float*)d_ws;
    float* kbuf = qbuf + (size_t)n * FTOT;

    // 1) zero output
    zero_out_kernel<<<(out_size + 255) / 256, 256, 0, stream>>>(out, out_size);

    // 2) q/k projection via fp32 WMMA: grid (rowTiles, 3) x block (32, 8)
    //    8 waves/block cover the 24 (colTile, q/k) combos in grid.y tiles.
    const int rowTiles = (n + 15) / 16;
    dim3 gA(rowTiles, 3, 1), bA(32, 8, 1);
    qk_project_kernel<<<gA, bA, 0, stream>>>(x, Wq, Wk, qbuf, kbuf, n);

    // 3) edge gather + reduce + scatter: one wave32 per pair, 8 waves/block
    const int wavesPerBlock = 8;
    dim3 gB((npairs + wavesPerBlock - 1) / wavesPerBlock, 1, 1), bB(32 * wavesPerBlock, 1, 1);
    edge_attn_kernel<<<gB, bB, 0, stream>>>(qbuf, kbuf, w_ij, sph, phi_r, phi_c,
                                            idx_i, idx_j, out, npairs);
}